// MogCriterion_41807211659435
// MI455X (gfx1250) — compile-verified
//
#include <hip/hip_runtime.h>
#include <math.h>
#include <float.h>
#include <limits.h>

// ---------------- problem constants (from reference) ----------------
#define BATCH 16
#define A_TOTAL 34125
#define NLEV 6
#define V0 0.1f
#define V1 0.2f
// BOUNDS = {0, 25600, 32000, 33600, 34000, 34100, 34125}

#define BLK 256
#define MAXPOS 192   // cap positives tracked per level (data: <= 8*19)
#define MAXU 64      // cap unique boxes per level
#define NB_LOSS 1024 // stage-2 partial blocks

typedef float v2f __attribute__((ext_vector_type(2)));
typedef float v8f __attribute__((ext_vector_type(8)));

// -------- wave32 sum via V_WMMA_F32_16X16X4_F32 (A = ones) ----------
// B(4x16) rows K0/K1 carry the 32 lane values; D[m][n] = v_n + v_{n+16}
// for every m, so c[0] across lanes 0..31 = s_0..s_15, s_0..s_15.
// Four xor-shuffle adds finish the 16-way sum. Fixed order => deterministic.
__device__ __forceinline__ float wave_sum32(float v) {
#if __has_builtin(__builtin_amdgcn_wmma_f32_16x16x4_f32)
  v2f a; a[0] = 1.0f; a[1] = 1.0f;       // A = ones (16x4)
  v2f b; b[0] = v;    b[1] = 0.0f;       // B rows K0,K1 = lane values
  v8f c = {};
  c = __builtin_amdgcn_wmma_f32_16x16x4_f32(false, a, false, b,
                                            (short)0, c, false, false);
  float x = c[0];
  x += __shfl_xor(x, 1, 32);
  x += __shfl_xor(x, 2, 32);
  x += __shfl_xor(x, 4, 32);
  x += __shfl_xor(x, 8, 32);
  return x;
#else
  for (int off = 16; off > 0; off >>= 1) v += __shfl_xor(v, off, 32);
  return v;
#endif
}

// ---------------- stage 0: extract labels ----------------
__global__ void mog_init_kernel(const float* __restrict__ bt,
                                float* __restrict__ labels, int N) {
  for (int i = blockIdx.x * blockDim.x + threadIdx.x; i < N;
       i += gridDim.x * blockDim.x) {
    labels[i] = bt[(size_t)i * 5 + 4];
  }
}

// ---------------- stage 1: relabel (one block per batch) ----------------
// Relabeled anchors: labels <- 2.0f, coords stored in ovr[anchor*4..].
__global__ __launch_bounds__(BLK) void mog_relabel_kernel(
    const float* __restrict__ conf,      // (B, A, 1)
    const float* __restrict__ bt0,       // (B, A, 5) original (read-only)
    const float* __restrict__ anchors,   // (A, 4) x1y1x2y2
    float* __restrict__ ovr,             // (B, A, 4) sparse overrides
    float* __restrict__ labels) {        // (B, A)
  const int b = blockIdx.x;
  const int tid = threadIdx.x;
  const size_t base = (size_t)b * A_TOTAL;

  __shared__ int s_poscnt[NLEV];
  __shared__ int s_posidx[NLEV * MAXPOS];
  __shared__ int s_nu[5];
  __shared__ int s_ukey[5 * MAXU * 4];
  __shared__ float s_ubox[5 * MAXU * 4];
  __shared__ int s_ucnt[5 * MAXU];
  __shared__ float s_rd[BLK];
  __shared__ int s_ri[BLK];
  __shared__ int sh_cand[MAXU];
  __shared__ float sh_gt[4];
  __shared__ int sh_s, sh_e, sh_maxnum, sh_need, sh_skip;
  __shared__ float sh_tmpw;
  __shared__ int sh_start, sh_end;

  if (tid < NLEV) s_poscnt[tid] = 0;
  if (tid == 0) { sh_tmpw = 160.0f; sh_start = -25600; sh_end = 0; }
  __syncthreads();

  // --- collect positive anchor indices per level (from ORIGINAL bt) ---
  for (int a = tid; a < A_TOTAL; a += BLK) {
    if (bt0[(base + a) * 5 + 4] == 1.0f) {
      int lev = (a < 25600) ? 0 : (a < 32000) ? 1 : (a < 33600) ? 2
              : (a < 34000) ? 3 : (a < 34100) ? 4 : 5;
      int p = atomicAdd(&s_poscnt[lev], 1);
      if (p < MAXPOS) s_posidx[lev * MAXPOS + p] = a;
    }
  }
  __syncthreads();

  // --- per-level decode + dedupe + lexicographic sort (levels 0..4) ---
  if (tid < 5) {
    const int li = tid;
    const int n = min(s_poscnt[li], MAXPOS);
    int nu = 0;
    for (int j = 0; j < n; ++j) {
      int a = s_posidx[li * MAXPOS + j];
      const float* ap = anchors + (size_t)a * 4;
      float cx = (ap[0] + ap[2]) * 0.5f, cy = (ap[1] + ap[3]) * 0.5f;
      float w = ap[2] - ap[0], h = ap[3] - ap[1];
      const float* dp = bt0 + (base + a) * 5;
      float ncx = cx + dp[0] * V0 * w;
      float ncy = cy + dp[1] * V0 * h;
      float nw = w * expf(dp[2] * V1);
      float nh = h * expf(dp[3] * V1);
      float box[4] = {ncx - nw * 0.5f, ncy - nh * 0.5f,
                      ncx + nw * 0.5f, ncy + nh * 0.5f};
      int key[4];
      float rb[4];
      for (int k = 0; k < 4; ++k) {
        key[k] = (int)rintf(box[k] * 1000.0f);   // np.round(.,3) semantics
        rb[k] = (float)key[k] / 1000.0f;
      }
      int fnd = -1;
      for (int t2 = 0; t2 < nu; ++t2) {
        const int* kk = &s_ukey[(li * MAXU + t2) * 4];
        if (kk[0] == key[0] && kk[1] == key[1] && kk[2] == key[2] &&
            kk[3] == key[3]) { fnd = t2; break; }
      }
      if (fnd >= 0) {
        s_ucnt[li * MAXU + fnd]++;
      } else if (nu < MAXU) {
        int u = li * MAXU + nu;
        for (int k = 0; k < 4; ++k) { s_ukey[u * 4 + k] = key[k]; s_ubox[u * 4 + k] = rb[k]; }
        s_ucnt[u] = 1;
        ++nu;
      }
    }
    // stable lexicographic insertion sort (np.unique row order)
    for (int i = 1; i < nu; ++i) {
      int tk[4]; float tb[4];
      int u = li * MAXU + i;
      for (int k = 0; k < 4; ++k) { tk[k] = s_ukey[u * 4 + k]; tb[k] = s_ubox[u * 4 + k]; }
      int tc = s_ucnt[u];
      int j = i - 1;
      while (j >= 0) {
        int uj = li * MAXU + j;
        bool gt = false, eqrun = true;
        for (int k = 0; k < 4 && eqrun; ++k) {
          if (s_ukey[uj * 4 + k] > tk[k]) { gt = true; eqrun = false; }
          else if (s_ukey[uj * 4 + k] < tk[k]) { eqrun = false; }
        }
        if (!gt) break;
        for (int k = 0; k < 4; ++k) {
          s_ukey[(uj + 1) * 4 + k] = s_ukey[uj * 4 + k];
          s_ubox[(uj + 1) * 4 + k] = s_ubox[uj * 4 + k];
        }
        s_ucnt[uj + 1] = s_ucnt[uj];
        --j;
      }
      int up = li * MAXU + j + 1;
      for (int k = 0; k < 4; ++k) { s_ukey[up * 4 + k] = tk[k]; s_ubox[up * 4 + k] = tb[k]; }
      s_ucnt[up] = tc;
    }
    s_nu[li] = nu;
  }
  __syncthreads();

  // --- relabel loop: sequential over levels 0..4 and unique boxes ---
  for (int li = 0; li < 5; ++li) {
    __syncthreads();
    const int nu = s_nu[li];
    if (nu == 0) continue;                 // window does NOT advance (ref quirk)
    if (tid == 0) {
      int tw2 = (int)(sh_tmpw * sh_tmpw);
      sh_start += tw2; sh_end += tw2; sh_tmpw *= 0.5f;
      int mx = 0;
      for (int t = 0; t < nu; ++t) mx = max(mx, s_ucnt[li * MAXU + t]);
      sh_maxnum = min(mx, MAXU);
      sh_s = sh_start; sh_e = sh_end;
    }
    __syncthreads();
    const int s = sh_s, e = sh_e, maxnum = sh_maxnum;

    for (int ti = 0; ti < nu; ++ti) {
      __syncthreads();
      if (tid == 0) {
        int cnt = s_ucnt[li * MAXU + ti];
        sh_skip = (cnt >= maxnum) ? 1 : 0;
        sh_need = maxnum - cnt;
        for (int k = 0; k < 4; ++k) sh_gt[k] = s_ubox[(li * MAXU + ti) * 4 + k];
      }
      __syncthreads();
      if (sh_skip) continue;

      const float gcx = (sh_gt[0] + sh_gt[2]) * 0.5f;
      const float gcy = (sh_gt[1] + sh_gt[3]) * 0.5f;

      // top-maxnum nearest anchors (stable argsort semantics: repeated
      // argmin with smallest-index tie-break)
      for (int sel = 0; sel < maxnum; ++sel) {
        float bd = FLT_MAX;
        int bi = INT_MAX;
        for (int i = s + tid; i < e; i += BLK) {
          bool ex = false;
          for (int j = 0; j < sel; ++j)
            if (sh_cand[j] == i) { ex = true; break; }
          if (ex) continue;
          const float* ap = anchors + (size_t)i * 4;
          float dx = (ap[0] + ap[2]) * 0.5f - gcx;
          float dy = (ap[1] + ap[3]) * 0.5f - gcy;
          float d = dx * dx + dy * dy;     // monotone in hypot
          if (d < bd) { bd = d; bi = i; }  // ascending scan => first index on tie
        }
        s_rd[tid] = bd; s_ri[tid] = bi;
        __syncthreads();
        for (int off = BLK / 2; off > 0; off >>= 1) {
          if (tid < off) {
            float od = s_rd[tid + off]; int oi = s_ri[tid + off];
            if (od < s_rd[tid] || (od == s_rd[tid] && oi < s_ri[tid])) {
              s_rd[tid] = od; s_ri[tid] = oi;
            }
          }
          __syncthreads();
        }
        if (tid == 0) sh_cand[sel] = (s_ri[0] == INT_MAX) ? -1 : s_ri[0];
        __syncthreads();
      }

      if (tid == 0) {
        // scores (raw logits: sigmoid is monotone, same ordering/ties)
        float sc[MAXU];
        int ord[MAXU];
        for (int j = 0; j < maxnum; ++j) {
          int a = sh_cand[j];
          sc[j] = (a >= 0) ? conf[base + a] : -FLT_MAX;
          ord[j] = j;
        }
        // stable ascending insertion sort
        for (int i = 1; i < maxnum; ++i) {
          int o = ord[i]; float k = sc[o]; int j = i - 1;
          while (j >= 0 && sc[ord[j]] > k) { ord[j + 1] = ord[j]; --j; }
          ord[j + 1] = o;
        }
        const int need = sh_need;
        for (int j = maxnum - need; j < maxnum; ++j) {
          int a = sh_cand[ord[j]];
          if (a < 0) continue;
          size_t la = base + a;
          float lab = labels[la];
          if (lab == 1.0f || lab == 2.0f) continue;  // ref: rel[labels != 1]
          const float* ap = anchors + (size_t)a * 4;
          float cx = (ap[0] + ap[2]) * 0.5f, cy = (ap[1] + ap[3]) * 0.5f;
          float w = ap[2] - ap[0], h = ap[3] - ap[1];
          float gw = sh_gt[2] - sh_gt[0], gh = sh_gt[3] - sh_gt[1];
          float* d = ovr + la * 4;
          d[0] = (gcx - cx) / (w * V0);
          d[1] = (gcy - cy) / (h * V0);
          d[2] = logf(fmaxf(gw, 1e-6f) / w) / V1;
          d[3] = logf(fmaxf(gh, 1e-6f) / h) / V1;
          labels[la] = 2.0f;                // positive, coords in ovr
        }
      }
    }
  }
}

// ---------------- stage 2: streaming loss partials ----------------
// Traffic: conf + labels dense; loc/bt/ovr fetched only for positives.
__global__ __launch_bounds__(BLK) void mog_loss_partial_kernel(
    const float* __restrict__ conf, const float* __restrict__ loc,
    const float* __restrict__ bt0, const float* __restrict__ ovr,
    const float* __restrict__ labels, float* __restrict__ partials, int N) {
  float sb = 0.0f, sl = 0.0f, sp = 0.0f;
  for (int i = blockIdx.x * blockDim.x + threadIdx.x; i < N;
       i += gridDim.x * blockDim.x) {
    float z = conf[i];
    float lab = labels[i];
    float t = (lab == 1.0f || lab == 2.0f) ? 1.0f : 0.0f;
    sb += fmaxf(z, 0.0f) - z * t + log1pf(expf(-fabsf(z)));
    sp += t;
    if (t != 0.0f) {
      const float4 lv = *(const float4*)(loc + (size_t)i * 4);  // b128
      const float* bp = (lab == 2.0f) ? (ovr + (size_t)i * 4)
                                      : (bt0 + (size_t)i * 5);
      float d0 = lv.x - bp[0], d1 = lv.y - bp[1];
      float d2 = lv.z - bp[2], d3 = lv.w - bp[3];
      float a0 = fabsf(d0), a1 = fabsf(d1), a2 = fabsf(d2), a3 = fabsf(d3);
      sl += (a0 < 1.0f) ? 0.5f * d0 * d0 : a0 - 0.5f;
      sl += (a1 < 1.0f) ? 0.5f * d1 * d1 : a1 - 0.5f;
      sl += (a2 < 1.0f) ? 0.5f * d2 * d2 : a2 - 0.5f;
      sl += (a3 < 1.0f) ? 0.5f * d3 * d3 : a3 - 0.5f;
    }
  }
  __shared__ float sA[BLK], sB[BLK], sC[BLK];
  int tid = threadIdx.x;
  sA[tid] = sb; sB[tid] = sl; sC[tid] = sp;
  __syncthreads();
  for (int off = BLK / 2; off > 0; off >>= 1) {
    if (tid < off) {
      sA[tid] += sA[tid + off];
      sB[tid] += sB[tid + off];
      sC[tid] += sC[tid + off];
    }
    __syncthreads();
  }
  if (tid == 0) {
    partials[blockIdx.x] = sA[0];
    partials[NB_LOSS + blockIdx.x] = sB[0];
    partials[2 * NB_LOSS + blockIdx.x] = sC[0];
  }
}

// ---------------- stage 3: final reduce (WMMA wave fold) ----------------
__global__ __launch_bounds__(BLK) void mog_finalize_kernel(
    const float* __restrict__ partials, int nb, float* __restrict__ out) {
  __shared__ float sA[BLK], sB[BLK], sC[BLK];
  const int tid = threadIdx.x;
  float a = 0.0f, b = 0.0f, c = 0.0f;
  for (int i = tid; i < nb; i += BLK) {
    a += partials[i];
    b += partials[nb + i];
    c += partials[2 * nb + i];
  }
  sA[tid] = a; sB[tid] = b; sC[tid] = c;
  __syncthreads();
  for (int off = BLK / 2; off >= 32; off >>= 1) {
    if (tid < off) {
      sA[tid] += sA[tid + off];
      sB[tid] += sB[tid + off];
      sC[tid] += sC[tid + off];
    }
    __syncthreads();
  }
  // every wave holds the same 32 partials => EXEC all-ones for WMMA
  float ta = wave_sum32(sA[tid & 31]);
  float tb = wave_sum32(sB[tid & 31]);
  float tc = wave_sum32(sC[tid & 31]);
  if (tid == 0) {
    float npos = fmaxf(tc, 1.0f);
    float cls = 1.0f * ta / npos;   // CLS_W
    float lcl = 2.0f * tb / npos;   // LOC_W
    out[0] = cls; out[1] = lcl; out[2] = cls + lcl;
  }
}

// ---------------- launcher ----------------
extern "C" void kernel_launch(void* const* d_in, const int* in_sizes, int n_in,
                              void* d_out, int out_size, void* d_ws,
                              size_t ws_size, hipStream_t stream) {
  (void)in_sizes; (void)n_in; (void)out_size; (void)ws_size;
  const float* conf = (const float*)d_in[0];   // (16, 34125, 1)
  const float* loc = (const float*)d_in[1];    // (16, 34125, 4)
  const float* bt = (const float*)d_in[2];     // (16, 34125, 5)
  const float* anch = (const float*)d_in[3];   // (34125, 4)
  float* out = (float*)d_out;                  // 3 floats

  float* ws = (float*)d_ws;
  const int N = BATCH * A_TOTAL;
  float* labels = ws;                               // N floats
  float* ovr = labels + (size_t)N;                  // N*4 floats (sparse)
  float* partials = ovr + (size_t)N * 4;            // 3*NB_LOSS floats

  mog_init_kernel<<<1024, BLK, 0, stream>>>(bt, labels, N);
  mog_relabel_kernel<<<BATCH, BLK, 0, stream>>>(conf, bt, anch, ovr, labels);
  mog_loss_partial_kernel<<<NB_LOSS, BLK, 0, stream>>>(conf, loc, bt, ovr,
                                                       labels, partials, N);
  mog_finalize_kernel<<<1, BLK, 0, stream>>>(partials, NB_LOSS, out);
}